// ResidualBlock2_0_36910948942381
// MI455X (gfx1250) — compile-verified
//
#include <hip/hip_runtime.h>

// ---------------------------------------------------------------------------
// AdderNet residual block for MI455X (gfx1250, wave32).
// The L1-distance "conv" is not bilinear -> cannot use WMMA for the main
// FLOPs; it is VALU-bound (~4.8 G fp32 ops vs ~60 MB HBM traffic).
// WMMA (V_WMMA_F32_16X16X4_F32, ones-matrix trick) is used for the exact
// fp32 per-channel BatchNorm reductions.
// ---------------------------------------------------------------------------

typedef float v2f __attribute__((ext_vector_type(2)));
typedef float v8f __attribute__((ext_vector_type(8)));

#define B_   32
#define C_   64
#define H_   32
#define W_   32
#define KD   576   // C_ * 3 * 3
#define COG  16    // output channels per block
#define TILE 16    // spatial tile
#define CCH  32    // input-channel chunk staged in LDS

// --- transpose weights: w[co][k] -> wt[coGroup][k][coInGroup] (contiguous
// 16-float runs -> wave-uniform scalar loads in the hot loop) ---------------
__global__ __launch_bounds__(256) void wtrans_kernel(const float* __restrict__ w,
                                                     float* __restrict__ wt) {
  int idx = blockIdx.x * 256 + threadIdx.x;
  if (idx < C_ * KD) {
    int co = idx / KD, k = idx - co * KD;
    int cg = co >> 4, c = co & 15;
    wt[cg * (KD * COG) + k * COG + c] = w[idx];
  }
}

// --- adder2d: y[n,co,h,w] = -sum |x_patch - w[co]| --------------------------
__global__ __launch_bounds__(256) void adder_kernel(const float* __restrict__ in,
                                                    const float* __restrict__ wt,
                                                    float* __restrict__ y) {
  __shared__ float xs[CCH][TILE + 2][TILE + 2];   // 41,472 B
  const int tid    = threadIdx.x;
  const int cg     = blockIdx.x;        // 0..3 : group of 16 Cout
  const int tileId = blockIdx.y;        // 0..3 : 16x16 spatial tile
  const int n      = blockIdx.z;        // 0..31
  const int ty0 = (tileId >> 1) * TILE;
  const int tx0 = (tileId & 1) * TILE;
  const int tx = tid & 15, ty = tid >> 4;

  const float* wblk = wt + cg * (KD * COG);
  __builtin_prefetch(wblk, 0, 1);       // global_prefetch_b8 on weight block

  float acc[COG];
#pragma unroll
  for (int c = 0; c < COG; ++c) acc[c] = 0.f;

  for (int c0 = 0; c0 < C_; c0 += CCH) {
    // stage halo tile for CCH input channels into LDS (zero-padded border)
    for (int i = tid; i < CCH * (TILE + 2) * (TILE + 2); i += 256) {
      int ci = i / ((TILE + 2) * (TILE + 2));
      int r  = i - ci * ((TILE + 2) * (TILE + 2));
      int yy = r / (TILE + 2);
      int xx = r - yy * (TILE + 2);
      int gy = ty0 + yy - 1;
      int gx = tx0 + xx - 1;
      float v = 0.f;
      if ((unsigned)gy < (unsigned)H_ && (unsigned)gx < (unsigned)W_)
        v = in[(((size_t)n * C_ + (c0 + ci)) * H_ + gy) * W_ + gx];
      xs[ci][yy][xx] = v;
    }
    __syncthreads();

    for (int ci = 0; ci < CCH; ++ci) {
#pragma unroll
      for (int kh = 0; kh < 3; ++kh) {
#pragma unroll
        for (int kw = 0; kw < 3; ++kw) {
          const float xv = xs[ci][ty + kh][tx + kw];        // 1 ds broadcastable read
          const float* wk = wblk + (((c0 + ci) * 3 + kh) * 3 + kw) * COG; // uniform -> s_load
#pragma unroll
          for (int c = 0; c < COG; ++c)
            acc[c] += fabsf(xv - wk[c]);                    // v_sub + v_add(|src|)
        }
      }
    }
    __syncthreads();
  }

  const int gy = ty0 + ty, gx = tx0 + tx;
#pragma unroll
  for (int c = 0; c < COG; ++c)
    y[(((size_t)n * C_ + (cg * COG + c)) * H_ + gy) * W_ + gx] = -acc[c];
}

// --- BatchNorm statistics via WMMA ones-reduction ---------------------------
// One block per channel. D = ones(16x4) * B(4x16) + C accumulates column sums
// of each 64-element f32 tile in the matrix unit; row 0 (+row 8) of D holds
// the running column sums, so a 32-lane reduce of acc[0] gives 2x the total.
__global__ __launch_bounds__(256) void stats_kernel(const float* __restrict__ y,
                                                    const float* __restrict__ g,
                                                    const float* __restrict__ b,
                                                    float* __restrict__ ss) {
  const int c    = blockIdx.x;          // channel 0..63
  const int tid  = threadIdx.x;
  const int wave = tid >> 5, lane = tid & 31;

  v8f accS = {};
  v8f accQ = {};
  v2f ones = {1.f, 1.f};

  const float* chan = y + (size_t)c * (H_ * W_);
  // 32768 elements per channel, 8 waves, 64 elements (2/lane) per WMMA issue
  for (int base = wave * 64; base < B_ * H_ * W_; base += 8 * 64) {
    int e0 = base + lane * 2;                 // contiguous pair within channel
    int n0 = e0 >> 10, i0 = e0 & 1023;
    const float* p = chan + (size_t)n0 * (C_ * H_ * W_) + i0;
    v2f d = {p[0], p[1]};
    v2f q = {d.x * d.x, d.y * d.y};
    accS = __builtin_amdgcn_wmma_f32_16x16x4_f32(false, ones, false, d,
                                                 (short)0, accS, false, false);
    accQ = __builtin_amdgcn_wmma_f32_16x16x4_f32(false, ones, false, q,
                                                 (short)0, accQ, false, false);
  }

  float s = accS[0], q = accQ[0];
  for (int off = 16; off > 0; off >>= 1) {
    s += __shfl_xor(s, off, 32);
    q += __shfl_xor(q, off, 32);
  }
  s *= 0.5f;  // lanes 0-15 = row M=0, lanes 16-31 = row M=8 (same sums)
  q *= 0.5f;

  __shared__ float ls[8], lq[8];
  if (lane == 0) { ls[wave] = s; lq[wave] = q; }
  __syncthreads();
  if (tid == 0) {
    float S = 0.f, Q = 0.f;
    for (int i = 0; i < 8; ++i) { S += ls[i]; Q += lq[i]; }
    const float inv_n = 1.0f / (float)(B_ * H_ * W_);
    float mu    = S * inv_n;
    float var   = Q * inv_n - mu * mu;
    float inv   = rsqrtf(var + 1e-5f);
    float scale = g[c] * inv;
    ss[c]       = scale;                 // y*scale + shift == gamma*(y-mu)*rsqrt+beta
    ss[C_ + c]  = b[c] - mu * scale;
  }
}

// --- normalize (+ optional residual) + ReLU ---------------------------------
__global__ __launch_bounds__(256) void apply_kernel(const float* __restrict__ y,
                                                    const float* __restrict__ ss,
                                                    const float* __restrict__ resid,
                                                    float* __restrict__ out, int n) {
  int idx = blockIdx.x * 256 + threadIdx.x;
  if (idx < n) {
    int c = (idx >> 10) & (C_ - 1);
    float v = y[idx] * ss[c] + ss[C_ + c];
    if (resid) v += resid[idx];
    out[idx] = fmaxf(v, 0.f);
  }
}

extern "C" void kernel_launch(void* const* d_in, const int* in_sizes, int n_in,
                              void* d_out, int out_size, void* d_ws, size_t ws_size,
                              hipStream_t stream) {
  (void)in_sizes; (void)n_in; (void)out_size; (void)ws_size;
  const float* x  = (const float*)d_in[0];
  const float* w1 = (const float*)d_in[1];
  const float* g1 = (const float*)d_in[2];
  const float* b1 = (const float*)d_in[3];
  const float* w2 = (const float*)d_in[4];
  const float* g2 = (const float*)d_in[5];
  const float* b2 = (const float*)d_in[6];
  float* out = (float*)d_out;

  char* ws = (char*)d_ws;
  const size_t TEN   = (size_t)B_ * C_ * H_ * W_;    // 2,097,152 elements
  const size_t WTSZ  = (size_t)C_ * KD * sizeof(float); // 147,456 B
  float* ybuf = (float*)(ws);
  float* abuf = (float*)(ws + TEN * 4);
  float* wt1  = (float*)(ws + 2 * TEN * 4);
  float* wt2  = (float*)(ws + 2 * TEN * 4 + WTSZ);
  float* ss1  = (float*)(ws + 2 * TEN * 4 + 2 * WTSZ);
  float* ss2  = (float*)(ws + 2 * TEN * 4 + 2 * WTSZ + 512);

  wtrans_kernel<<<144, 256, 0, stream>>>(w1, wt1);
  wtrans_kernel<<<144, 256, 0, stream>>>(w2, wt2);

  dim3 agrid(C_ / COG, (H_ / TILE) * (W_ / TILE), B_);   // (4, 4, 32)

  // layer 1: adder -> BN(train stats) -> relu
  adder_kernel<<<agrid, 256, 0, stream>>>(x, wt1, ybuf);
  stats_kernel<<<C_, 256, 0, stream>>>(ybuf, g1, b1, ss1);
  apply_kernel<<<(int)(TEN / 256), 256, 0, stream>>>(ybuf, ss1, nullptr, abuf, (int)TEN);

  // layer 2: adder -> BN -> +x -> relu
  adder_kernel<<<agrid, 256, 0, stream>>>(abuf, wt2, ybuf);
  stats_kernel<<<C_, 256, 0, stream>>>(ybuf, g2, b2, ss2);
  apply_kernel<<<(int)(TEN / 256), 256, 0, stream>>>(ybuf, ss2, x, out, (int)TEN);
}